// AttLoRAModule_85839216378078
// MI455X (gfx1250) — compile-verified
//
#include <hip/hip_runtime.h>
#include <hip/hip_bf16.h>
#include <math.h>

// Problem constants (match reference)
#define BB   4
#define SS   2048
#define NIN  4096
#define NOUT 4096
#define NE   8
#define NR   64
#define NK   128
#define ER   (NE * NR)        // 512 = concatenated LoRA rank dim
#define SCALEMULT 1.0f        // SCALE * MULTIPLIER

// Tiling
#define BM  128
#define BN  128
#define BKC 32                // K-chunk staged in LDS
#define LDP 36                // padded LDS row stride (floats): 144B, 16B-aligned

typedef float v2f __attribute__((ext_vector_type(2)));
typedef float v8f __attribute__((ext_vector_type(8)));
typedef int   v4i __attribute__((ext_vector_type(4)));

static __device__ __forceinline__ v8f wmma_f32(v2f a, v2f b, v8f c) {
    // V_WMMA_F32_16X16X4_F32 : D = A(16x4,f32) * B(4x16,f32) + C(16x16,f32)
    return __builtin_amdgcn_wmma_f32_16x16x4_f32(
        /*neg_a=*/false, a, /*neg_b=*/false, b,
        /*c_mod=*/(short)0, c, /*reuse_a=*/false, /*reuse_b=*/false);
}

// ---------------------------------------------------------------------------
// CDNA5 async global->LDS copy (ASYNCcnt-tracked), with sync fallback.
// Probe round 2 showed: builtin exists, 4 args, param0 = v4i in addrspace(1).
// ---------------------------------------------------------------------------
typedef __attribute__((address_space(1))) void as1_void;
typedef __attribute__((address_space(3))) void as3_void;
typedef __attribute__((address_space(1))) v4i  as1_v4i;
typedef __attribute__((address_space(3))) v4i  as3_v4i;

#if __has_builtin(__builtin_amdgcn_global_load_async_to_lds_b128)
#define HAVE_ASYNC_LDS 1
static __device__ __forceinline__ void g2l_b128(const float* g, float* l) {
    __builtin_amdgcn_global_load_async_to_lds_b128(
        (as1_v4i*)(as1_void*)(float*)g,
        (as3_v4i*)(as3_void*)l,
        /*offset=*/0, /*cpol=*/0);
}
#else
#define HAVE_ASYNC_LDS 0
static __device__ __forceinline__ void g2l_b128(const float* g, float* l) {
    *(float4*)l = *(const float4*)g;   // synchronous fallback
}
#endif

#if HAVE_ASYNC_LDS
#  if __has_builtin(__builtin_amdgcn_s_wait_asynccnt)
#    define WAIT_ASYNC(n) __builtin_amdgcn_s_wait_asynccnt(n)
#  else
#    define WAIT_ASYNC(n) asm volatile("s_wait_asynccnt %0" :: "i"(n) : "memory")
#  endif
#else
#  define WAIT_ASYNC(n) ((void)0)
#endif

// ---------------------------------------------------------------------------
// Kernel 1: xmean[b,i] = mean_s x[b,s,i]   (split over S, atomic accumulate)
// grid: (B*NIN/256, SCHUNKS) ; xmean must be pre-zeroed
// ---------------------------------------------------------------------------
#define SCHUNKS 8
__global__ __launch_bounds__(256) void k_colmean(const float* __restrict__ x,
                                                 float* __restrict__ xmean) {
    const int idx = blockIdx.x * 256 + threadIdx.x;   // 0 .. B*NIN-1
    const int b = idx / NIN;
    const int i = idx - b * NIN;
    const int s0 = blockIdx.y * (SS / SCHUNKS);
    const float* p = x + (size_t)b * SS * NIN + (size_t)s0 * NIN + i;
    float acc = 0.0f;
    for (int s = 0; s < SS / SCHUNKS; ++s)
        acc += p[(size_t)s * NIN];
    atomicAdd(&xmean[idx], acc * (1.0f / (float)SS));
}

// ---------------------------------------------------------------------------
// Kernel 2: router. q = xmean @ Wq^T ; scores = q.keys/sqrt(K) ; softmax -> coeffs
// single block, 128 threads
// ---------------------------------------------------------------------------
__global__ __launch_bounds__(128) void k_router(const float* __restrict__ xmean,
                                                const float* __restrict__ Wq,
                                                const float* __restrict__ keys,
                                                float* __restrict__ coeffs) {
    __shared__ float qs[BB * NK];
    __shared__ float sc[BB * NE];
    const int t = threadIdx.x;          // 0..127 == k index
    for (int b = 0; b < BB; ++b) {
        const float* xm = xmean + b * NIN;
        const float* w  = Wq + (size_t)t * NIN;
        float acc = 0.0f;
        for (int i = 0; i < NIN; ++i) acc += xm[i] * w[i];
        qs[b * NK + t] = acc;
    }
    __syncthreads();
    if (t < BB * NE) {
        const int b = t / NE, e = t - b * NE;
        float s = 0.0f;
        const float* q = qs + b * NK;
        const float* ke = keys + e * NK;
        for (int k = 0; k < NK; ++k) s += q[k] * ke[k];
        sc[t] = s * (1.0f / sqrtf((float)NK));
    }
    __syncthreads();
    if (t < BB) {
        float m = -1e30f;
        for (int e = 0; e < NE; ++e) m = fmaxf(m, sc[t * NE + e]);
        float ex[NE], sum = 0.0f;
        for (int e = 0; e < NE; ++e) { ex[e] = __expf(sc[t * NE + e] - m); sum += ex[e]; }
        const float inv = 1.0f / sum;
        for (int e = 0; e < NE; ++e) coeffs[t * NE + e] = ex[e] * inv;
    }
}

// ---------------------------------------------------------------------------
// Kernel 3: H[b,s,c] = coeffs[b, c/64] * (x[b] @ down)[s,c]   c in [0,512)
// A tile staged via async global->LDS; B tile is a transpose gather (sync).
// grid: (ER/BN, SS/BM, B), 256 threads (8 waves), wave tile 32x64
// ---------------------------------------------------------------------------
__global__ __launch_bounds__(256) void k_downproj(const float* __restrict__ x,
                                                  const float* __restrict__ down,
                                                  const float* __restrict__ coeffs,
                                                  float* __restrict__ H) {
    __shared__ float As[BM * LDP];
    __shared__ float Bs[BN * LDP];
    const int b  = blockIdx.z;
    const int s0 = blockIdx.y * BM;
    const int c0 = blockIdx.x * BN;
    const int tid  = threadIdx.x;
    const int lane = tid & 31, wid = tid >> 5;
    const int waveM = (wid & 3) * 32;
    const int waveN = (wid >> 2) * 64;
    const int lr = lane & 15;
    const int lh = lane >> 4;

    const v8f vzero = {0.f,0.f,0.f,0.f,0.f,0.f,0.f,0.f};
    v8f acc[2][4];
#pragma unroll
    for (int i = 0; i < 2; ++i)
#pragma unroll
        for (int j = 0; j < 4; ++j) acc[i][j] = vzero;

    const float* xb = x + (size_t)b * SS * NIN;

    for (int k0 = 0; k0 < NIN; k0 += BKC) {
        // A tile (async): As[m][k] = x[b, s0+m, k0+k]
#pragma unroll
        for (int r = 0; r < 4; ++r) {
            const int t = tid + 256 * r;
            const int row = t >> 3, cg = (t & 7) << 2;
            g2l_b128(&xb[(size_t)(s0 + row) * NIN + k0 + cg], &As[row * LDP + cg]);
        }
        // B tile (sync transpose): Bs[c][k] = down[e, k0+k, r]
#pragma unroll
        for (int r = 0; r < 4; ++r) {
            const int t = tid + 256 * r;
            const int k = t >> 5, cg = (t & 31) << 2;
            const int c = c0 + cg;
            const int e = c >> 6, rr = c & 63;
            const float4 v = *(const float4*)&down[((size_t)e * NIN + k0 + k) * NR + rr];
            Bs[(cg + 0) * LDP + k] = v.x;
            Bs[(cg + 1) * LDP + k] = v.y;
            Bs[(cg + 2) * LDP + k] = v.z;
            Bs[(cg + 3) * LDP + k] = v.w;
        }
        WAIT_ASYNC(0);
        __syncthreads();
#pragma unroll
        for (int kk = 0; kk < BKC; kk += 4) {
            const int kf = kk + (lh << 1);
            const v2f a0 = *(const v2f*)&As[(waveM +  0 + lr) * LDP + kf];
            const v2f a1 = *(const v2f*)&As[(waveM + 16 + lr) * LDP + kf];
            const v2f b0 = *(const v2f*)&Bs[(waveN +  0 + lr) * LDP + kf];
            const v2f b1 = *(const v2f*)&Bs[(waveN + 16 + lr) * LDP + kf];
            const v2f b2 = *(const v2f*)&Bs[(waveN + 32 + lr) * LDP + kf];
            const v2f b3 = *(const v2f*)&Bs[(waveN + 48 + lr) * LDP + kf];
            acc[0][0] = wmma_f32(a0, b0, acc[0][0]);
            acc[0][1] = wmma_f32(a0, b1, acc[0][1]);
            acc[0][2] = wmma_f32(a0, b2, acc[0][2]);
            acc[0][3] = wmma_f32(a0, b3, acc[0][3]);
            acc[1][0] = wmma_f32(a1, b0, acc[1][0]);
            acc[1][1] = wmma_f32(a1, b1, acc[1][1]);
            acc[1][2] = wmma_f32(a1, b2, acc[1][2]);
            acc[1][3] = wmma_f32(a1, b3, acc[1][3]);
        }
        __syncthreads();
    }

    float* Hb = H + (size_t)b * SS * ER;
#pragma unroll
    for (int i = 0; i < 2; ++i) {
#pragma unroll
        for (int j = 0; j < 4; ++j) {
            const int col = c0 + waveN + j * 16 + lr;
            const int e   = col >> 6;
            const float scale = coeffs[b * NE + e] * SCALEMULT;
            const int rbase = s0 + waveM + i * 16 + lh * 8;
#pragma unroll
            for (int v = 0; v < 8; ++v)
                Hb[(size_t)(rbase + v) * ER + col] = acc[i][j][v] * scale;
        }
    }
}

// ---------------------------------------------------------------------------
// Kernel 4: out[b,s,o] = x[b] @ W_org^T + H[b] @ up    (fused accumulate)
// Phase 1 double-buffered with async global->LDS (A and B layout-preserving).
// grid: (NOUT/BN, SS/BM, B), 256 threads
// ---------------------------------------------------------------------------
__global__ __launch_bounds__(256) void k_main_gemm(const float* __restrict__ x,
                                                   const float* __restrict__ Worg,
                                                   const float* __restrict__ H,
                                                   const float* __restrict__ up,
                                                   float* __restrict__ out) {
    __shared__ float As[2][BM * LDP];
    __shared__ float Bs[2][BN * LDP];
    const int b  = blockIdx.z;
    const int s0 = blockIdx.y * BM;
    const int o0 = blockIdx.x * BN;
    const int tid  = threadIdx.x;
    const int lane = tid & 31, wid = tid >> 5;
    const int waveM = (wid & 3) * 32;
    const int waveN = (wid >> 2) * 64;
    const int lr = lane & 15;
    const int lh = lane >> 4;

    const v8f vzero = {0.f,0.f,0.f,0.f,0.f,0.f,0.f,0.f};
    v8f acc[2][4];
#pragma unroll
    for (int i = 0; i < 2; ++i)
#pragma unroll
        for (int j = 0; j < 4; ++j) acc[i][j] = vzero;

    const float* xb = x + (size_t)b * SS * NIN;
    const float* Hb = H + (size_t)b * SS * ER;

    // stage helpers (async issue; completion via WAIT_ASYNC + barrier)
    auto stageA1 = [&](int buf, int k0) {
#pragma unroll
        for (int r = 0; r < 4; ++r) {
            const int t = tid + 256 * r;
            const int row = t >> 3, cg = (t & 7) << 2;
            g2l_b128(&xb[(size_t)(s0 + row) * NIN + k0 + cg], &As[buf][row * LDP + cg]);
        }
    };
    auto stageB1 = [&](int buf, int k0) {
#pragma unroll
        for (int r = 0; r < 4; ++r) {
            const int t = tid + 256 * r;
            const int row = t >> 3, cg = (t & 7) << 2;
            g2l_b128(&Worg[(size_t)(o0 + row) * NIN + k0 + cg], &Bs[buf][row * LDP + cg]);
        }
    };
    auto stageA2 = [&](int buf, int k0) {
#pragma unroll
        for (int r = 0; r < 4; ++r) {
            const int t = tid + 256 * r;
            const int row = t >> 3, cg = (t & 7) << 2;
            g2l_b128(&Hb[(size_t)(s0 + row) * ER + k0 + cg], &As[buf][row * LDP + cg]);
        }
    };
    auto compute = [&](int buf) {
#pragma unroll
        for (int kk = 0; kk < BKC; kk += 4) {
            const int kf = kk + (lh << 1);
            const v2f a0 = *(const v2f*)&As[buf][(waveM +  0 + lr) * LDP + kf];
            const v2f a1 = *(const v2f*)&As[buf][(waveM + 16 + lr) * LDP + kf];
            const v2f b0 = *(const v2f*)&Bs[buf][(waveN +  0 + lr) * LDP + kf];
            const v2f b1 = *(const v2f*)&Bs[buf][(waveN + 16 + lr) * LDP + kf];
            const v2f b2 = *(const v2f*)&Bs[buf][(waveN + 32 + lr) * LDP + kf];
            const v2f b3 = *(const v2f*)&Bs[buf][(waveN + 48 + lr) * LDP + kf];
            acc[0][0] = wmma_f32(a0, b0, acc[0][0]);
            acc[0][1] = wmma_f32(a0, b1, acc[0][1]);
            acc[0][2] = wmma_f32(a0, b2, acc[0][2]);
            acc[0][3] = wmma_f32(a0, b3, acc[0][3]);
            acc[1][0] = wmma_f32(a1, b0, acc[1][0]);
            acc[1][1] = wmma_f32(a1, b1, acc[1][1]);
            acc[1][2] = wmma_f32(a1, b2, acc[1][2]);
            acc[1][3] = wmma_f32(a1, b3, acc[1][3]);
        }
    };

    // ---- Phase 1: base path, K = NIN, double-buffered async staging
    stageA1(0, 0);
    stageB1(0, 0);
    for (int c = 0; c < NIN / BKC; ++c) {
        const int cur = c & 1;
        const int k0n = (c + 1) * BKC;
        if (k0n < NIN) {
            stageA1(cur ^ 1, k0n);
            stageB1(cur ^ 1, k0n);
            WAIT_ASYNC(8);          // 8 newer asyncs outstanding -> chunk c complete
        } else {
            WAIT_ASYNC(0);
        }
        __syncthreads();
        compute(cur);
        __syncthreads();
    }

    // ---- Phase 2: LoRA path, K = ER(512); A async double-buffered, B sync transpose
    stageA2(0, 0);
    for (int c = 0; c < ER / BKC; ++c) {
        const int cur = c & 1;
        const int k0  = c * BKC;
        const int k0n = k0 + BKC;
        if (k0n < ER) stageA2(cur ^ 1, k0n);
        // Bs[cur][n][k] = up[(k0+k)*NOUT + o0+n]  (sync gather+transpose)
#pragma unroll
        for (int r = 0; r < 4; ++r) {
            const int t = tid + 256 * r;
            const int k = t >> 5, ng = (t & 31) << 2;
            const float4 v = *(const float4*)&up[(size_t)(k0 + k) * NOUT + o0 + ng];
            Bs[cur][(ng + 0) * LDP + k] = v.x;
            Bs[cur][(ng + 1) * LDP + k] = v.y;
            Bs[cur][(ng + 2) * LDP + k] = v.z;
            Bs[cur][(ng + 3) * LDP + k] = v.w;
        }
        if (k0n < ER) { WAIT_ASYNC(4); } else { WAIT_ASYNC(0); }
        __syncthreads();
        compute(cur);
        __syncthreads();
    }

    // ---- Store (C/D layout: VGPR v -> M = v + 8*(lane>=16), N = lane%16)
    float* ob = out + (size_t)b * SS * NOUT;
#pragma unroll
    for (int i = 0; i < 2; ++i) {
#pragma unroll
        for (int j = 0; j < 4; ++j) {
            const int col   = o0 + waveN + j * 16 + lr;
            const int rbase = s0 + waveM + i * 16 + lh * 8;
#pragma unroll
            for (int v = 0; v < 8; ++v)
                ob[(size_t)(rbase + v) * NOUT + col] = acc[i][j][v];
        }
    }
}

// ---------------------------------------------------------------------------
// Launch wiring
// Inputs (order): x, W_org, lora_down, lora_up, lora_keys, Wq  (all fp32)
// Workspace layout: [0,128)=coeffs ; [256, 256+64K)=xmean ; [1MB, 1MB+16MB)=H
// ---------------------------------------------------------------------------
extern "C" void kernel_launch(void* const* d_in, const int* in_sizes, int n_in,
                              void* d_out, int out_size, void* d_ws, size_t ws_size,
                              hipStream_t stream) {
    const float* x    = (const float*)d_in[0];
    const float* Worg = (const float*)d_in[1];
    const float* down = (const float*)d_in[2];
    const float* up   = (const float*)d_in[3];
    const float* keys = (const float*)d_in[4];
    const float* Wq   = (const float*)d_in[5];
    float* out = (float*)d_out;

    char* ws = (char*)d_ws;
    float* coeffs = (float*)(ws + 0);
    float* xmean  = (float*)(ws + 256);
    float* H      = (float*)(ws + (1u << 20));

    // zero xmean accumulator (atomic reduction target)
    (void)hipMemsetAsync(xmean, 0, (size_t)BB * NIN * sizeof(float), stream);

    // 1) column mean of x
    k_colmean<<<dim3(BB * NIN / 256, SCHUNKS), 256, 0, stream>>>(x, xmean);

    // 2) router -> coeffs
    k_router<<<1, 128, 0, stream>>>(xmean, Wq, keys, coeffs);

    // 3) down-projection, coeff-scaled: H[b,s,512]
    k_downproj<<<dim3(ER / BN, SS / BM, BB), 256, 0, stream>>>(x, down, coeffs, H);

    // 4) fused base GEMM + LoRA up-projection
    k_main_gemm<<<dim3(NOUT / BN, SS / BM, BB), 256, 0, stream>>>(x, Worg, H, up, out);
}